// WB_CLAHE_55224689492257
// MI455X (gfx1250) — compile-verified
//
#include <hip/hip_runtime.h>
#include <stdint.h>

typedef float v2f __attribute__((ext_vector_type(2)));
typedef float v8f __attribute__((ext_vector_type(8)));

#define IMGPIX 262144u                 // 512*512
#define NPIX   (16u * IMGPIX)
#define EPS_LAB 0.0088564516790356308f // 216/24389
#define KAPPA_LAB 903.2962962962963f   // 24389/27

// ---------- helpers ----------
__device__ __forceinline__ uint32_t fkey(float f) {
  uint32_t u = __float_as_uint(f);
  return (u & 0x80000000u) ? ~u : (u | 0x80000000u);
}
__device__ __forceinline__ float unkey(uint32_t k) {
  uint32_t u = (k & 0x80000000u) ? (k & 0x7FFFFFFFu) : ~k;
  return __uint_as_float(u);
}
__device__ __forceinline__ float srgb2lin(float c) {
  return (c <= 0.04045f) ? c * (1.0f / 12.92f)
                         : powf((c + 0.055f) / 1.055f, 2.4f);
}
__device__ __forceinline__ float lin2srgb(float c) {
  c = fmaxf(c, 0.0f);
  return (c <= 0.0031308f) ? c * 12.92f
                           : 1.055f * powf(c, 1.0f / 2.4f) - 0.055f;
}
__device__ __forceinline__ float finv_lab(float f) {
  float f3 = f * f * f;
  return (f3 > EPS_LAB) ? f3 : (116.0f * f - 16.0f) / KAPPA_LAB;
}
__device__ __forceinline__ void lab2rgb(float l255, float a, float b,
                                        float& r, float& g, float& bl) {
  float L  = l255 * (100.0f / 255.0f);
  float fy = (L + 16.0f) / 116.0f;
  float fx = fy + a / 500.0f;
  float fz = fy - b / 200.0f;
  float X = finv_lab(fx) * 0.95047f;
  float Y = finv_lab(fy);
  float Z = finv_lab(fz) * 1.08883f;
  r  = fminf(fmaxf(lin2srgb( 3.2404542f * X - 1.5371385f * Y - 0.4985314f * Z), 0.f), 1.f);
  g  = fminf(fmaxf(lin2srgb(-0.969266f  * X + 1.8760108f * Y + 0.041556f  * Z), 0.f), 1.f);
  bl = fminf(fmaxf(lin2srgb( 0.0556434f * X - 0.2040259f * Y + 1.0572252f * Z), 0.f), 1.f);
}

// ---------- K0: init hist + min/max keys ----------
__global__ void k_init(uint32_t* hist, uint32_t* minKey, uint32_t* maxKey) {
  uint32_t t = blockIdx.x * 256u + threadIdx.x;
  if (t < 266240u) hist[t] = 0u;                         // 16*256 + 16*64*256
  else if (t < 266256u) minKey[t - 266240u] = 0xFFFFFFFFu;
  else if (t < 266272u) maxKey[t - 266256u] = 0u;
}

// ---------- K1: RGB -> Lab, per-image L min/max ----------
__global__ void k_rgb2lab(const float* __restrict__ x, float* __restrict__ wsL,
                          float* __restrict__ wsA, float* __restrict__ wsB,
                          uint32_t* minKey, uint32_t* maxKey) {
  __shared__ float smin[8], smax[8];
  uint32_t idx = blockIdx.x * 256u + threadIdx.x;
  uint32_t n = idx >> 18;
  uint32_t rem = idx & (IMGPIX - 1u);
  const float* px = x + (size_t)n * 3u * IMGPIX + rem;
  float lr = srgb2lin(px[0]);
  float lg = srgb2lin(px[IMGPIX]);
  float lb = srgb2lin(px[2u * IMGPIX]);
  float X = (0.4124564f * lr + 0.3575761f * lg + 0.1804375f * lb) / 0.95047f;
  float Y = (0.2126729f * lr + 0.7151522f * lg + 0.072175f  * lb);
  float Z = (0.0193339f * lr + 0.119192f  * lg + 0.9503041f * lb) / 1.08883f;
  float fx = (X > EPS_LAB) ? cbrtf(X) : (KAPPA_LAB * X + 16.0f) / 116.0f;
  float fy = (Y > EPS_LAB) ? cbrtf(Y) : (KAPPA_LAB * Y + 16.0f) / 116.0f;
  float fz = (Z > EPS_LAB) ? cbrtf(Z) : (KAPPA_LAB * Z + 16.0f) / 116.0f;
  float L = 116.0f * fy - 16.0f;
  wsL[idx] = L;
  wsA[idx] = 500.0f * (fx - fy);
  wsB[idx] = 200.0f * (fy - fz);

  float vmin = L, vmax = L;
  for (int o = 16; o > 0; o >>= 1) {
    vmin = fminf(vmin, __shfl_xor(vmin, o, 32));
    vmax = fmaxf(vmax, __shfl_xor(vmax, o, 32));
  }
  int lane = threadIdx.x & 31, wid = threadIdx.x >> 5;
  if (lane == 0) { smin[wid] = vmin; smax[wid] = vmax; }
  __syncthreads();
  if (threadIdx.x == 0) {
    float m = smin[0], M = smax[0];
    for (int w = 1; w < 8; w++) { m = fminf(m, smin[w]); M = fmaxf(M, smax[w]); }
    atomicMin(&minKey[n], fkey(m));
    atomicMax(&maxKey[n], fkey(M));
  }
}

// ---------- K2: quantize + tile/image histograms (1 block = 1 tile) ----------
__global__ void k_hist(float* __restrict__ wsL, const uint32_t* minKey,
                       const uint32_t* maxKey, uint32_t* imgHist,
                       uint32_t* tileHist) {
  __shared__ uint32_t sh[256];
  int n = blockIdx.x >> 6;
  int tile = blockIdx.x & 63;
  int ty = tile >> 3, tx = tile & 7;
  int tid = threadIdx.x;
  sh[tid] = 0u;
  __syncthreads();
  float lmin = unkey(minKey[n]);
  float denom = unkey(maxKey[n]) - lmin + 1e-6f;
#pragma unroll
  for (int k = 0; k < 16; k++) {
    int p = tid + k * 256;
    int ly = p >> 6, lx = p & 63;
    uint32_t idx = (uint32_t)n * IMGPIX + (uint32_t)(ty * 64 + ly) * 512u + (uint32_t)(tx * 64 + lx);
    float L = wsL[idx];
    float l255 = (L - lmin) / denom * 255.0f;
    int v = (int)fminf(fmaxf(rintf(l255), 0.0f), 255.0f);
    wsL[idx] = __int_as_float(v);   // store l_int bit-cast in place
    atomicAdd(&sh[v], 1u);
  }
  __syncthreads();
  uint32_t c = sh[tid];
  tileHist[(size_t)blockIdx.x * 256 + tid] = c;
  atomicAdd(&imgHist[n * 256 + tid], c);
}

// ---------- K3: build LUTs — WMMA-based 256-bin prefix sum, 1 wave/LUT ----------
__global__ void k_luts(const uint32_t* __restrict__ imgHist,
                       const uint32_t* __restrict__ tileHist,
                       float* __restrict__ lutEq, float* __restrict__ lutClahe) {
  __shared__ float hsh[256];
  __shared__ float rs[16];
  int task = blockIdx.x;              // 0..15 equalize, 16..1039 CLAHE tiles
  int lane = threadIdx.x;             // wave32, one wave per block
  bool clahe = task >= 16;
  const uint32_t* src = clahe ? (tileHist + (size_t)(task - 16) * 256)
                              : (imgHist + (size_t)task * 256);
  float e = 0.0f;
#pragma unroll
  for (int i = 0; i < 8; i++) {
    int j = lane * 8 + i;
    float c = (float)src[j];
    if (clahe) { float cc = fminf(c, 32.0f); e += c - cc; c = cc; }
    hsh[j] = c;
  }
  for (int o = 16; o > 0; o >>= 1) e += __shfl_xor(e, o, 32);
  float add = clahe ? e * (1.0f / 256.0f) : 0.0f;
  __syncthreads();

  int half = lane >> 4;     // 0: lanes 0-15, 1: lanes 16-31
  int mcol = lane & 15;     // A row index / B,C,D column index
  v2f A[4], B[4];
#pragma unroll
  for (int c4 = 0; c4 < 4; c4++) {
    int k = 4 * c4 + 2 * half;          // global K index per ISA 16x4 layout
    A[c4].x = hsh[mcol * 16 + k]     + add;
    A[c4].y = hsh[mcol * 16 + k + 1] + add;
    B[c4].x = (k     <= mcol) ? 1.0f : 0.0f;  // upper-triangular ones
    B[c4].y = (k + 1 <= mcol) ? 1.0f : 0.0f;
  }
  v8f acc = {};
  acc = __builtin_amdgcn_wmma_f32_16x16x4_f32(false, A[0], false, B[0], (short)0, acc, false, false);
  acc = __builtin_amdgcn_wmma_f32_16x16x4_f32(false, A[1], false, B[1], (short)0, acc, false, false);
  acc = __builtin_amdgcn_wmma_f32_16x16x4_f32(false, A[2], false, B[2], (short)0, acc, false, false);
  acc = __builtin_amdgcn_wmma_f32_16x16x4_f32(false, A[3], false, B[3], (short)0, acc, false, false);

  // row totals (P[m][15]) live in lanes 15 / 31
  if (lane == 15 || lane == 31) {
#pragma unroll
    for (int r = 0; r < 8; r++) rs[half * 8 + r] = acc[r];
  }
  __syncthreads();
  float pre[16];
  {
    float p = 0.0f;
#pragma unroll
    for (int m = 0; m < 16; m++) { pre[m] = p; p += rs[m]; }
  }
  float scale = clahe ? (255.0f / 4096.0f) : (255.0f / 262144.0f);
  float* outp = clahe ? (lutClahe + (size_t)(task - 16) * 256)
                      : (lutEq + (size_t)task * 256);
#pragma unroll
  for (int r = 0; r < 8; r++) {
    int m = r + half * 8;
    outp[m * 16 + mcol] = rintf((acc[r] + pre[m]) * scale);
  }
}

// ---------- K4: LUT apply + CLAHE bilinear + Lab->RGB x2 ----------
__global__ void k_final(const float* __restrict__ lintF, const float* __restrict__ wsA,
                        const float* __restrict__ wsB, const float* __restrict__ lutEq,
                        const float* __restrict__ lutClahe, float* __restrict__ out) {
  uint32_t idx = blockIdx.x * 256u + threadIdx.x;
  uint32_t n = idx >> 18;
  uint32_t rem = idx & (IMGPIX - 1u);
  int y = (int)(rem >> 9), xc = (int)(rem & 511u);
  int v = __float_as_int(lintF[idx]);
  float a = wsA[idx], b = wsB[idx];
  float lEq = lutEq[n * 256 + v];

  float cy = fminf(fmaxf((y + 0.5f) / 64.0f - 0.5f, 0.0f), 7.0f);
  float cx = fminf(fmaxf((xc + 0.5f) / 64.0f - 0.5f, 0.0f), 7.0f);
  int y0 = min((int)floorf(cy), 6);
  int x0 = min((int)floorf(cx), 6);
  float wy = cy - (float)y0, wx = cx - (float)x0;
  const float* lc = lutClahe + (size_t)n * 64 * 256 + v;
  float v00 = lc[(y0 * 8 + x0) * 256];
  float v01 = lc[(y0 * 8 + x0 + 1) * 256];
  float v10 = lc[((y0 + 1) * 8 + x0) * 256];
  float v11 = lc[((y0 + 1) * 8 + x0 + 1) * 256];
  float top = v00 * (1.0f - wx) + v01 * wx;
  float bot = v10 * (1.0f - wx) + v11 * wx;
  float lCl = top * (1.0f - wy) + bot * wy;

  float r, g, bl;
  size_t b1 = (size_t)n * 3u * IMGPIX + rem;
  lab2rgb(lEq, a, b, r, g, bl);
  out[b1] = 255.0f * r; out[b1 + IMGPIX] = 255.0f * g; out[b1 + 2u * IMGPIX] = 255.0f * bl;
  lab2rgb(lCl, a, b, r, g, bl);
  size_t b2 = b1 + (size_t)16u * 3u * IMGPIX;
  out[b2] = 255.0f * r; out[b2 + IMGPIX] = 255.0f * g; out[b2 + 2u * IMGPIX] = 255.0f * bl;
}

// ---------- host ----------
extern "C" void kernel_launch(void* const* d_in, const int* in_sizes, int n_in,
                              void* d_out, int out_size, void* d_ws, size_t ws_size,
                              hipStream_t stream) {
  (void)in_sizes; (void)n_in; (void)out_size; (void)ws_size;
  const float* x = (const float*)d_in[0];
  float* out = (float*)d_out;

  float* wsL = (float*)d_ws;
  float* wsA = wsL + NPIX;
  float* wsB = wsA + NPIX;
  uint32_t* imgHist  = (uint32_t*)(wsB + NPIX);        // 16*256
  uint32_t* tileHist = imgHist + 16 * 256;             // 16*64*256 (contiguous w/ imgHist)
  uint32_t* minKey   = tileHist + 16 * 64 * 256;       // 16
  uint32_t* maxKey   = minKey + 16;                    // 16
  float* lutEq    = (float*)(maxKey + 16);             // 16*256
  float* lutClahe = lutEq + 16 * 256;                  // 16*64*256

  k_init<<<(266272u + 255u) / 256u, 256, 0, stream>>>(imgHist, minKey, maxKey);
  k_rgb2lab<<<NPIX / 256u, 256, 0, stream>>>(x, wsL, wsA, wsB, minKey, maxKey);
  k_hist<<<16 * 64, 256, 0, stream>>>(wsL, minKey, maxKey, imgHist, tileHist);
  k_luts<<<16 + 16 * 64, 32, 0, stream>>>(imgHist, tileHist, lutEq, lutClahe);
  k_final<<<NPIX / 256u, 256, 0, stream>>>(wsL, wsA, wsB, lutEq, lutClahe, out);
}